// DistGCNGrad_29575144800295
// MI455X (gfx1250) — compile-verified
//
#include <hip/hip_runtime.h>
#include <math.h>

// ---------------------------------------------------------------------------
// DistGCN forward on gfx1250 (MI455X): two GCN layers + log_softmax.
// GEMMs use v_wmma_f32_16x16x4_f32 (exact f32) with pre-transposed, padded
// weights so the K-loop is branch-free (b64 loads for both A and B frags),
// loads batched ahead of the WMMA group for staggered s_wait_loadcnt.
// Edge aggregation uses native f32 global atomics (h1+agg1 ~= 205MB working
// set, mostly L2-resident on the 192MB L2, avg ~18 touches per node row).
// ---------------------------------------------------------------------------

typedef __attribute__((ext_vector_type(2))) float v2f;
typedef __attribute__((ext_vector_type(8))) float v8f;

#define WAVE 32

// ---------------- degree / normalization ----------------

__global__ __launch_bounds__(256) void k_init_deg(float* __restrict__ deg, int n) {
  int i = blockIdx.x * 256 + threadIdx.x;
  if (i < n) deg[i] = 1.0f;  // self loop
}

__global__ __launch_bounds__(256) void k_accum_deg(const int* __restrict__ dst, int e,
                                                   float* __restrict__ deg) {
  int i = blockIdx.x * 256 + threadIdx.x;
  if (i < e) unsafeAtomicAdd(&deg[dst[i]], 1.0f);
}

__global__ __launch_bounds__(256) void k_finalize_deg(const float* __restrict__ deg,
                                                      float* __restrict__ isd,
                                                      float* __restrict__ invd, int n) {
  int i = blockIdx.x * 256 + threadIdx.x;
  if (i < n) {
    float d = deg[i];
    isd[i]  = rsqrtf(d);
    invd[i] = 1.0f / d;
  }
}

// ---------------- weight transpose + zero-pad: Wt[n*K+k] = W[k*Nsrc+n] ------

__global__ __launch_bounds__(256) void k_transpose_pad(const float* __restrict__ W,
                                                       float* __restrict__ Wt,
                                                       int K, int Nsrc, int Npad) {
  int idx = blockIdx.x * 256 + threadIdx.x;
  if (idx < K * Npad) {
    int n = idx / K;
    int k = idx % K;
    Wt[idx] = (n < Nsrc) ? W[(size_t)k * Nsrc + n] : 0.0f;
  }
}

// ---------------- WMMA GEMM: H[M x Nvalid] = act(A)[M x Kd] @ W[Kd x N] -----
// Wt is the transposed, padded weight: Wt[col * Kd + k]. One wave owns one
// 16-row M tile and sweeps NT_PER_WAVE 16-col N tiles, reusing the A fragment.
// Per K-step: issue A + all B loads first, then the WMMA group, so waits are
// staggered and loads pipeline against the matrix ops.
// f32 WMMA VGPR layout: lanes 0-15 hold K={k,k+1}, lanes 16-31 K={k+2,k+3};
// D rows: M = half*8 + r, col = lane&15.

template <int Kd, int NTILES, int GROUPS, bool RELU_IN>
__global__ __launch_bounds__(256) void k_gemm_wmma(const float* __restrict__ A,
                                                   const float* __restrict__ Wt,
                                                   float* __restrict__ H,
                                                   int M, int Nvalid) {
  constexpr int NT_PER_WAVE = NTILES / GROUPS;
  int gw = blockIdx.x * (256 / WAVE) + (threadIdx.x >> 5);
  int mtiles = M >> 4;
  if (gw >= mtiles * GROUPS) return;           // whole-wave uniform: EXEC stays full
  int tileM = gw / GROUPS;
  int tnb   = (gw % GROUPS) * NT_PER_WAVE;

  int lane = threadIdx.x & (WAVE - 1);
  int half = lane >> 4;
  int mr   = lane & 15;
  int koff = half << 1;                        // this lane's K sub-offset {0,2}

  const float* Arow = A + (size_t)(tileM * 16 + mr) * Kd + koff;
  const float* Bcol[NT_PER_WAVE];
#pragma unroll
  for (int t = 0; t < NT_PER_WAVE; t++)
    Bcol[t] = Wt + (size_t)((tnb + t) * 16 + mr) * Kd + koff;

  v8f acc[NT_PER_WAVE] = {};

  for (int k = 0; k < Kd; k += 4) {
    v2f av = *(const v2f*)(Arow + k);
    v2f bv[NT_PER_WAVE];
#pragma unroll
    for (int t = 0; t < NT_PER_WAVE; t++)
      bv[t] = *(const v2f*)(Bcol[t] + k);
    if (RELU_IN) { av.x = fmaxf(av.x, 0.0f); av.y = fmaxf(av.y, 0.0f); }
#pragma unroll
    for (int t = 0; t < NT_PER_WAVE; t++)
      acc[t] = __builtin_amdgcn_wmma_f32_16x16x4_f32(
          /*neg_a=*/false, av, /*neg_b=*/false, bv[t],
          /*c_mod=*/(short)0, acc[t], /*reuse_a=*/false, /*reuse_b=*/false);
  }

#pragma unroll
  for (int t = 0; t < NT_PER_WAVE; t++) {
    int col = (tnb + t) * 16 + mr;
    if (col < Nvalid) {
      int rowbase = tileM * 16 + (half << 3);
#pragma unroll
      for (int r = 0; r < 8; r++)
        H[(size_t)(rowbase + r) * Nvalid + col] = acc[t][r];
    }
  }
}

// ---------------- agg init: agg[i][f] = h[i][f]*invd[i] + bias[f] -----------

template <int F>
__global__ __launch_bounds__(256) void k_init_agg(const float* __restrict__ h,
                                                  const float* __restrict__ invd,
                                                  const float* __restrict__ bias,
                                                  float* __restrict__ agg, int n) {
  size_t i = (size_t)blockIdx.x * 256 + threadIdx.x;
  if (i < (size_t)n * F) {
    int node = (int)(i / F);
    int f    = (int)(i % F);
    agg[i] = h[i] * invd[node] + bias[f];
  }
}

// ---------------- edge scatter: agg[dst] += h[src] * isd[src]*isd[dst] ------
// One wave per edge; lanes stride the feature dimension (coalesced 128B reads,
// native f32 atomic adds -- L2-resident, this is the roofline-dominant phase).

template <int F>
__global__ __launch_bounds__(256) void k_scatter(const int* __restrict__ src,
                                                 const int* __restrict__ dst, int e,
                                                 const float* __restrict__ isd,
                                                 const float* __restrict__ h,
                                                 float* __restrict__ agg) {
  int wi = blockIdx.x * (256 / WAVE) + (threadIdx.x >> 5);
  if (wi >= e) return;
  int lane = threadIdx.x & 31;
  int s = src[wi], d = dst[wi];
  float norm = isd[s] * isd[d];
  const float* hs = h + (size_t)s * F;
  float* ad = agg + (size_t)d * F;
#pragma unroll
  for (int f = lane; f < F; f += WAVE)
    unsafeAtomicAdd(&ad[f], hs[f] * norm);
}

// ---------------- log_softmax over 40 classes, one wave per row -------------

__global__ __launch_bounds__(256) void k_logsoftmax40(const float* __restrict__ a,
                                                      float* __restrict__ out, int n) {
  int row = blockIdx.x * (256 / WAVE) + (threadIdx.x >> 5);
  if (row >= n) return;
  int lane = threadIdx.x & 31;
  const float* ar = a + (size_t)row * 40;
  float v0 = ar[lane];                                   // lanes 0..31 all < 40
  float v1 = (lane < 8) ? ar[32 + lane] : -INFINITY;
  float m = fmaxf(v0, v1);
#pragma unroll
  for (int off = 16; off; off >>= 1) m = fmaxf(m, __shfl_xor(m, off, 32));
  float s = __expf(v0 - m) + ((lane < 8) ? __expf(v1 - m) : 0.0f);
#pragma unroll
  for (int off = 16; off; off >>= 1) s += __shfl_xor(s, off, 32);
  float lse = m + __logf(s);
  float* orow = out + (size_t)row * 40;
  orow[lane] = v0 - lse;
  if (lane < 8) orow[32 + lane] = v1 - lse;
}

// ---------------------------------------------------------------------------

extern "C" void kernel_launch(void* const* d_in, const int* in_sizes, int n_in,
                              void* d_out, int out_size, void* d_ws, size_t ws_size,
                              hipStream_t stream) {
  const float* x  = (const float*)d_in[0];
  const int*   le = (const int*)d_in[1];
  const int*   re = (const int*)d_in[2];
  const float* W1 = (const float*)d_in[3];
  const float* b1 = (const float*)d_in[4];
  const float* W2 = (const float*)d_in[5];
  const float* b2 = (const float*)d_in[6];

  const int FIN = 128, FHID = 256, FOUT = 40, FOUTP = 48;
  int N  = in_sizes[0] / FIN;
  int eL = in_sizes[1] / 2;
  int eR = in_sizes[2] / 2;
  const int* lsrc = le;      const int* ldst = le + eL;
  const int* rsrc = re;      const int* rdst = re + eR;

  // workspace carve-out (256B aligned)
  char* ws = (char*)d_ws;
  size_t off = 0;
  auto carve = [&](size_t bytes) {
    size_t o = off;
    off = (off + bytes + 255) & ~(size_t)255;
    return o;
  };
  float* deg  = (float*)(ws + carve((size_t)N * 4));
  float* isd  = (float*)(ws + carve((size_t)N * 4));
  float* invd = (float*)(ws + carve((size_t)N * 4));
  float* Wt1  = (float*)(ws + carve((size_t)FHID * FIN * 4));    // [256 x 128]
  float* Wt2  = (float*)(ws + carve((size_t)FOUTP * FHID * 4));  // [48 x 256], rows 40..47 zero
  float* bufA = (float*)(ws + carve((size_t)N * FHID * 4));      // h1, later h2
  float* bufB = (float*)(ws + carve((size_t)N * FHID * 4));      // agg1, later agg2
  float* h1 = bufA, *agg1 = bufB;
  float* h2 = bufA, *agg2 = bufB;
  float* outp = (float*)d_out;
  (void)ws_size; (void)n_in; (void)out_size;

  int nb = (N + 255) / 256;
  k_init_deg<<<nb, 256, 0, stream>>>(deg, N);
  k_accum_deg<<<(eL + 255) / 256, 256, 0, stream>>>(ldst, eL, deg);
  k_accum_deg<<<(eR + 255) / 256, 256, 0, stream>>>(rdst, eR, deg);
  k_finalize_deg<<<nb, 256, 0, stream>>>(deg, isd, invd, N);

  // transpose + pad weights once per launch (tiny)
  k_transpose_pad<<<(FIN * FHID + 255) / 256, 256, 0, stream>>>(W1, Wt1, FIN, FHID, FHID);
  k_transpose_pad<<<(FHID * FOUTP + 255) / 256, 256, 0, stream>>>(W2, Wt2, FHID, FOUT, FOUTP);

  // Layer 1: h1 = x @ W1   (K=128, 16 N-tiles, 8 tiles/wave -> 2 groups)
  {
    int jobs = (N / 16) * 2;
    k_gemm_wmma<128, 16, 2, false><<<(jobs + 7) / 8, 256, 0, stream>>>(x, Wt1, h1, N, FHID);
  }
  {
    size_t total = (size_t)N * FHID;
    k_init_agg<256><<<(int)((total + 255) / 256), 256, 0, stream>>>(h1, invd, b1, agg1, N);
  }
  k_scatter<256><<<(eL + 7) / 8, 256, 0, stream>>>(lsrc, ldst, eL, isd, h1, agg1);
  k_scatter<256><<<(eR + 7) / 8, 256, 0, stream>>>(rsrc, rdst, eR, isd, h1, agg1);

  // Layer 2: h2 = relu(agg1) @ W2   (K=256, 3 N-tiles padded to 48, store-guard 40)
  {
    int jobs = N / 16;
    k_gemm_wmma<256, 3, 1, true><<<(jobs + 7) / 8, 256, 0, stream>>>(agg1, Wt2, h2, N, FOUT);
  }
  {
    size_t total = (size_t)N * FOUT;
    k_init_agg<40><<<(int)((total + 255) / 256), 256, 0, stream>>>(h2, invd, b2, agg2, N);
  }
  k_scatter<40><<<(eL + 7) / 8, 256, 0, stream>>>(lsrc, ldst, eL, isd, h2, agg2);
  k_scatter<40><<<(eR + 7) / 8, 256, 0, stream>>>(rsrc, rdst, eR, isd, h2, agg2);

  k_logsoftmax40<<<(N + 7) / 8, 256, 0, stream>>>(agg2, outp, N);
}